// SparseMapSequences_90829968375935
// MI455X (gfx1250) — compile-verified
//
#include <hip/hip_runtime.h>
#include <hip/hip_bf16.h>

typedef float v2f __attribute__((ext_vector_type(2)));
typedef float v4f __attribute__((ext_vector_type(4)));
typedef float v8f __attribute__((ext_vector_type(8)));

#define TOTAL_ROWS 262144
#define NSEG 2048
#define DIM  256

// ---------------------------------------------------------------------------
// Kernel 1: per-segment mean pooling.  One block (256 thr = 8 waves) per
// segment.  Thread layout: rowgrp = tid/64 (0..3), col4 = (tid%64)*4.
// Each thread streams float4 column-chunks with a 4-row stride, then LDS
// reduces the 4 row-group partials down to one mean row of 256 floats.
// Reads are 16B-aligned, fully coalesced, non-temporal (touch-once data).
// ---------------------------------------------------------------------------
__global__ void __launch_bounds__(256)
seg_mean_kernel(const float* __restrict__ values,
                const int*   __restrict__ indices,   // (NSEG, 2) int32
                float*       __restrict__ seq_mean)  // (NSEG, DIM)
{
    __shared__ float sdata[4 * DIM];

    const int seg   = blockIdx.x;
    const int start = indices[2 * seg + 0];
    const int end   = indices[2 * seg + 1];
    const int len   = end - start;

    const int tid    = threadIdx.x;
    const int rowgrp = tid >> 6;          // 0..3
    const int col4   = (tid & 63) << 2;   // 0,4,...,252

    v4f acc = {0.f, 0.f, 0.f, 0.f};
    for (int r = start + rowgrp; r < end; r += 4) {
        const v4f* p = (const v4f*)(values + (size_t)r * DIM + col4);
        v4f v = __builtin_nontemporal_load(p);
        acc += v;
    }
    sdata[rowgrp * DIM + col4 + 0] = acc.x;
    sdata[rowgrp * DIM + col4 + 1] = acc.y;
    sdata[rowgrp * DIM + col4 + 2] = acc.z;
    sdata[rowgrp * DIM + col4 + 3] = acc.w;
    __syncthreads();

    // tid now owns one column
    float s = sdata[0 * DIM + tid] + sdata[1 * DIM + tid] +
              sdata[2 * DIM + tid] + sdata[3 * DIM + tid];
    seq_mean[(size_t)seg * DIM + tid] = s * (1.0f / (float)len);
}

// ---------------------------------------------------------------------------
// Kernel 2: out = seq_mean @ W + b  via V_WMMA_F32_16X16X4_F32 (full f32).
// 2048 output tiles of 16x16; one wave per tile; 64 chained k=4 WMMAs.
// Tile index t = ntile*128 + mtile so the 8 waves of a block share the same
// 16-column strip of W (L0/WGP$ reuse).
//
// VGPR layouts (ISA 7.12.2, 32-bit):
//   A 16x4 : lane l holds A[l&15][2*(l>>4) + {0,1}]  -> one float2 load
//   B 4x16 : VGPR v, lane l holds B[v + 2*(l>>4)][l&15]
//   C 16x16: VGPR v, lane l holds C[v + 8*(l>>4)][l&15]
// ---------------------------------------------------------------------------
__global__ void __launch_bounds__(256)
mean_gemm_wmma_kernel(const float* __restrict__ seq_mean, // (NSEG, DIM)
                      const float* __restrict__ W,        // (DIM, DIM)
                      const float* __restrict__ bias,     // (DIM)
                      float*       __restrict__ out)      // (NSEG, DIM)
{
    const int wave  = threadIdx.x >> 5;            // 0..7
    const int lane  = threadIdx.x & 31;
    const int tile  = blockIdx.x * 8 + wave;       // 0..2047
    const int mtile = tile & 127;                  // 128 M-tiles
    const int ntile = tile >> 7;                   // 16  N-tiles

    const int half = lane >> 4;                    // 0 or 1
    const int l15  = lane & 15;

    const int row0 = mtile * 16;
    const int col  = ntile * 16 + l15;

    const float* Arow  = seq_mean + (size_t)(row0 + l15) * DIM + 2 * half;
    const float* Bcol  = W + col;

    v8f c = {};
    #pragma unroll 4
    for (int k = 0; k < DIM; k += 4) {
        // A fragment: K = k + 2*half + {0,1}
        v2f a;
        {
            const float2 a2 = *(const float2*)(Arow + k);
            a.x = a2.x; a.y = a2.y;
        }
        // B fragment: VGPR v -> K = k + v + 2*half, N = l15
        v2f bfrag;
        bfrag.x = Bcol[(size_t)(k + 0 + 2 * half) * DIM];
        bfrag.y = Bcol[(size_t)(k + 1 + 2 * half) * DIM];

        c = __builtin_amdgcn_wmma_f32_16x16x4_f32(
                /*neg_a=*/false, a, /*neg_b=*/false, bfrag,
                /*c_mod=*/(short)0, c, /*reuse_a=*/false, /*reuse_b=*/false);
    }

    const float bv = bias[col];
    #pragma unroll
    for (int v = 0; v < 8; ++v) {
        const int row = row0 + v + 8 * half;
        out[(size_t)row * DIM + col] = c[v] + bv;
    }
}

// ---------------------------------------------------------------------------
extern "C" void kernel_launch(void* const* d_in, const int* in_sizes, int n_in,
                              void* d_out, int out_size, void* d_ws, size_t ws_size,
                              hipStream_t stream)
{
    const float* values  = (const float*)d_in[0];   // (262144, 256) f32
    const int*   indices = (const int*)  d_in[1];   // (2048, 2)     int32
    const float* W       = (const float*)d_in[2];   // (256, 256)    f32
    const float* bias    = (const float*)d_in[3];   // (256,)        f32
    float*       out     = (float*)d_out;           // (2048, 256)   f32

    float* seq_mean = (float*)d_ws;                 // 2048*256*4 = 2 MB scratch

    seg_mean_kernel<<<NSEG, 256, 0, stream>>>(values, indices, seq_mean);

    // 2048 tiles / 8 waves per block = 256 blocks
    mean_gemm_wmma_kernel<<<NSEG / 8, 256, 0, stream>>>(seq_mean, W, bias, out);
}